// TensorNetInteraction_75505525063696
// MI455X (gfx1250) — compile-verified
//
#include <hip/hip_runtime.h>
#include <hip/hip_bf16.h>

#define UNITS 64
#define NRBF  32

typedef float v2f __attribute__((ext_vector_type(2)));
typedef float v8f __attribute__((ext_vector_type(8)));

__device__ __forceinline__ v8f wmma_f32_4(v2f a, v2f b, v8f c) {
    // V_WMMA_F32_16X16X4_F32 : D = A(16x4 f32) * B(4x16 f32) + C(16x16 f32)
    return __builtin_amdgcn_wmma_f32_16x16x4_f32(false, a, false, b, (short)0, c,
                                                 false, false);
}

__device__ __forceinline__ float silu_f(float x) {
    return x / (1.0f + __expf(-x));
}

// ---------------------------------------------------------------------------
// Kernel 1: per (node,channel): normalize X, decompose into compressed I/A/S.
// P layout: 9 planes of [N*64]: 0:i0  1..3:a01,a02,a12  4..8:s00,s11,s01,s02,s12
// ---------------------------------------------------------------------------
__global__ void node_prep_kernel(const float* __restrict__ X,
                                 float* __restrict__ P, int NU) {
    int t = blockIdx.x * blockDim.x + threadIdx.x;
    if (t >= NU) return;
    const float* x = X + (size_t)t * 9;
    float v[9], nrm = 0.f;
#pragma unroll
    for (int i = 0; i < 9; ++i) { v[i] = x[i]; nrm += v[i] * v[i]; }
    float inv = 1.f / (nrm + 1.f);
#pragma unroll
    for (int i = 0; i < 9; ++i) v[i] *= inv;
    float i0  = (v[0] + v[4] + v[8]) * (1.f / 3.f);
    P[0 * NU + t] = i0;
    P[1 * NU + t] = 0.5f * (v[1] - v[3]);
    P[2 * NU + t] = 0.5f * (v[2] - v[6]);
    P[3 * NU + t] = 0.5f * (v[5] - v[7]);
    P[4 * NU + t] = v[0] - i0;
    P[5 * NU + t] = v[4] - i0;
    P[6 * NU + t] = 0.5f * (v[1] + v[3]);
    P[7 * NU + t] = 0.5f * (v[2] + v[6]);
    P[8 * NU + t] = 0.5f * (v[5] + v[7]);
}

// ---------------------------------------------------------------------------
// Kernel 2/7: channel-linear via WMMA f32 16x16x4, MT M-tiles per wave.
// Guard-free hot loop: caller guarantees all MT tiles are in range.
// Per plane k: O_k[n,d] = sum_c T_k[n,c] * W[d,c], W = W0(k=0), W1(k=1..3),
// W2(k=4..8).  grid = 9 * ngroups * 4.
// ---------------------------------------------------------------------------
template <int MT>
__global__ __launch_bounds__(32) void chanlin_kernel(
    const float* __restrict__ T, const float* __restrict__ W0,
    const float* __restrict__ W1, const float* __restrict__ W2,
    float* __restrict__ O, int Nn, int mtile0, int ngroups) {
    int tiles_per_plane = ngroups * 4;
    int gid   = blockIdx.x;
    int plane = gid / tiles_per_plane;
    int rem   = gid % tiles_per_plane;
    int g  = rem >> 2;
    int nt = rem & 3;
    const float* W = (plane == 0) ? W0 : (plane < 4 ? W1 : W2);
    int NU = Nn * UNITS;
    const float* Tp = T + (size_t)plane * NU;
    float* Op       = O + (size_t)plane * NU;

    int lane = threadIdx.x & 31;
    int lo = lane & 15, hi = lane >> 4;
    int m0 = (mtile0 + g * MT) * 16;
    const float* brow = W + (size_t)(nt * 16 + lo) * UNITS;

    v8f acc[MT];
#pragma unroll
    for (int mm = 0; mm < MT; ++mm) acc[mm] = (v8f){};

#pragma unroll
    for (int kb = 0; kb < UNITS; kb += 4) {
        int k0 = kb + hi * 2;
        v2f b;
        b.x = brow[k0]; b.y = brow[k0 + 1];
#pragma unroll
        for (int mm = 0; mm < MT; ++mm) {
            const float* ar = Tp + (size_t)(m0 + mm * 16 + lo) * UNITS;
            v2f a;
            a.x = ar[k0]; a.y = ar[k0 + 1];
            acc[mm] = wmma_f32_4(a, b, acc[mm]);
        }
    }
#pragma unroll
    for (int mm = 0; mm < MT; ++mm) {
#pragma unroll
        for (int i = 0; i < 8; ++i) {
            int m = m0 + mm * 16 + i + hi * 8;
            Op[(size_t)m * UNITS + nt * 16 + lo] = acc[mm][i];
        }
    }
}

// ---------------------------------------------------------------------------
// CSR build: histogram by src, one-block scan, scatter.
// ---------------------------------------------------------------------------
__global__ void zero_int_kernel(int* __restrict__ p, int n) {
    int i = blockIdx.x * blockDim.x + threadIdx.x;
    if (i < n) p[i] = 0;
}

__global__ void hist_kernel(const int* __restrict__ eidx, int* __restrict__ cnt,
                            int E) {
    int e = blockIdx.x * blockDim.x + threadIdx.x;
    if (e < E) atomicAdd(&cnt[eidx[e]], 1);   // src = edge_index[0]
}

__global__ __launch_bounds__(1024) void scan_kernel(
    const int* __restrict__ cnt, int* __restrict__ rowptr,
    int* __restrict__ cursor, int Nn) {
    __shared__ int buf[1024];
    __shared__ int carry;
    int tid = threadIdx.x;
    if (tid == 0) carry = 0;
    __syncthreads();
    for (int base = 0; base < Nn; base += 1024) {
        int i = base + tid;
        int v = (i < Nn) ? cnt[i] : 0;
        buf[tid] = v;
        __syncthreads();
        for (int off = 1; off < 1024; off <<= 1) {
            int add = (tid >= off) ? buf[tid - off] : 0;
            __syncthreads();
            buf[tid] += add;
            __syncthreads();
        }
        int incl  = buf[tid];
        int cbase = carry;
        if (i < Nn) {
            int excl = cbase + incl - v;
            rowptr[i] = excl;
            cursor[i] = excl;
        }
        __syncthreads();
        if (tid == 0) carry = cbase + buf[1023];
        __syncthreads();
    }
    if (tid == 0) rowptr[Nn] = carry;
}

__global__ void scatter_kernel(const int* __restrict__ eidx,
                               int* __restrict__ cursor,
                               int* __restrict__ elist, int E) {
    int e = blockIdx.x * blockDim.x + threadIdx.x;
    if (e < E) {
        int pos = atomicAdd(&cursor[eidx[e]], 1);
        elist[pos] = e;
    }
}

// ---------------------------------------------------------------------------
// Kernel 5: edge MLP via WMMA f32. One wave = 32 edges (2 M-tiles), so each
// weight B-fragment load feeds 2 WMMA chains (2x weight reuse + ILP-2).
// Writes ea = silu(mlp) * cutoff into EA[E,192].
// ---------------------------------------------------------------------------
__global__ __launch_bounds__(128) void edge_mlp_kernel(
    const float* __restrict__ ew, const float* __restrict__ attr,
    const float* __restrict__ Ws0, const float* __restrict__ bs0,
    const float* __restrict__ Ws1, const float* __restrict__ bs1,
    const float* __restrict__ Ws2, const float* __restrict__ bs2,
    float* __restrict__ EA, int E) {
    __shared__ float lds_h0[4][32 * 64];
    __shared__ float lds_h1[4][32 * 128];
    __shared__ float lds_c[4][32];

    int wave = threadIdx.x >> 5;
    int lane = threadIdx.x & 31;
    int lo = lane & 15, hi = lane >> 4;
    int e0 = (blockIdx.x * 4 + wave) * 32;
    if (e0 >= E) return;

    {
        float r = ew[e0 + lane];
        lds_c[wave][lane] =
            (r < 1.0f) ? 0.5f * (__cosf(3.14159265358979f * r) + 1.0f) : 0.0f;
    }

    // ---- layer 0: [32 x 32] @ Ws0^T -> [32 x 64], silu ----
    const float* ar0 = attr + (size_t)(e0 + lo) * NRBF;
    const float* ar1 = attr + (size_t)(e0 + 16 + lo) * NRBF;
#pragma unroll
    for (int nt = 0; nt < 4; ++nt) {
        const float* brow = Ws0 + (size_t)(nt * 16 + lo) * NRBF;
        v8f acc0 = {}, acc1 = {};
#pragma unroll
        for (int kb = 0; kb < NRBF; kb += 4) {
            int k0 = kb + hi * 2;
            v2f a0, a1, b;
            b.x  = brow[k0]; b.y  = brow[k0 + 1];
            a0.x = ar0[k0];  a0.y = ar0[k0 + 1];
            a1.x = ar1[k0];  a1.y = ar1[k0 + 1];
            acc0 = wmma_f32_4(a0, b, acc0);
            acc1 = wmma_f32_4(a1, b, acc1);
        }
        float bias = bs0[nt * 16 + lo];
#pragma unroll
        for (int i = 0; i < 8; ++i) {
            int m = i + hi * 8;
            lds_h0[wave][m * 64 + nt * 16 + lo]        = silu_f(acc0[i] + bias);
            lds_h0[wave][(m + 16) * 64 + nt * 16 + lo] = silu_f(acc1[i] + bias);
        }
    }

    // ---- layer 1: [32 x 64] @ Ws1^T -> [32 x 128], silu ----
    const float* hr0 = &lds_h0[wave][lo * 64];
    const float* hr1 = &lds_h0[wave][(16 + lo) * 64];
#pragma unroll
    for (int nt = 0; nt < 8; ++nt) {
        const float* brow = Ws1 + (size_t)(nt * 16 + lo) * 64;
        v8f acc0 = {}, acc1 = {};
#pragma unroll
        for (int kb = 0; kb < 64; kb += 4) {
            int k0 = kb + hi * 2;
            v2f a0, a1, b;
            b.x  = brow[k0]; b.y  = brow[k0 + 1];
            a0.x = hr0[k0];  a0.y = hr0[k0 + 1];
            a1.x = hr1[k0];  a1.y = hr1[k0 + 1];
            acc0 = wmma_f32_4(a0, b, acc0);
            acc1 = wmma_f32_4(a1, b, acc1);
        }
        float bias = bs1[nt * 16 + lo];
#pragma unroll
        for (int i = 0; i < 8; ++i) {
            int m = i + hi * 8;
            lds_h1[wave][m * 128 + nt * 16 + lo]        = silu_f(acc0[i] + bias);
            lds_h1[wave][(m + 16) * 128 + nt * 16 + lo] = silu_f(acc1[i] + bias);
        }
    }

    // ---- layer 2: [32 x 128] @ Ws2^T -> [32 x 192], silu, * cutoff -> EA ----
    const float* gr0 = &lds_h1[wave][lo * 128];
    const float* gr1 = &lds_h1[wave][(16 + lo) * 128];
#pragma unroll
    for (int nt = 0; nt < 12; ++nt) {
        const float* brow = Ws2 + (size_t)(nt * 16 + lo) * 128;
        v8f acc0 = {}, acc1 = {};
#pragma unroll
        for (int kb = 0; kb < 128; kb += 4) {
            int k0 = kb + hi * 2;
            v2f a0, a1, b;
            b.x  = brow[k0]; b.y  = brow[k0 + 1];
            a0.x = gr0[k0];  a0.y = gr0[k0 + 1];
            a1.x = gr1[k0];  a1.y = gr1[k0 + 1];
            acc0 = wmma_f32_4(a0, b, acc0);
            acc1 = wmma_f32_4(a1, b, acc1);
        }
        float bias = bs2[nt * 16 + lo];
#pragma unroll
        for (int i = 0; i < 8; ++i) {
            int m = i + hi * 8;
            EA[(size_t)(e0 + m) * 192 + nt * 16 + lo] =
                silu_f(acc0[i] + bias) * lds_c[wave][m];
            EA[(size_t)(e0 + m + 16) * 192 + nt * 16 + lo] =
                silu_f(acc1[i] + bias) * lds_c[wave][m + 16];
        }
    }
}

// ---------------------------------------------------------------------------
// Kernel 6: message gather (atomic-free). One thread per (node, channel);
// the per-node edge list is wave-uniform -> scalar control flow, L2 gathers.
// ---------------------------------------------------------------------------
__global__ void message_gather_kernel(const int* __restrict__ eidx,
                                      const int* __restrict__ rowptr,
                                      const int* __restrict__ elist,
                                      const float* __restrict__ EA,
                                      const float* __restrict__ Q,
                                      float* __restrict__ Mg, int E, int NU) {
    int t = blockIdx.x * blockDim.x + threadIdx.x;
    if (t >= NU) return;
    int n = t >> 6, c = t & 63;
    float acc[9];
#pragma unroll
    for (int k = 0; k < 9; ++k) acc[k] = 0.f;
    int beg = rowptr[n], end = rowptr[n + 1];
    for (int j = beg; j < end; ++j) {
        int e   = elist[j];
        int dst = eidx[E + e];                 // edge_index[1]
        const float* ea = EA + (size_t)e * 192 + 3 * c;
        float e0v = ea[0], e1v = ea[1], e2v = ea[2];
        int bq = dst * UNITS + c;
        acc[0] += e0v * Q[0 * NU + bq];
#pragma unroll
        for (int k = 1; k < 4; ++k) acc[k] += e1v * Q[k * NU + bq];
#pragma unroll
        for (int k = 4; k < 9; ++k) acc[k] += e2v * Q[k * NU + bq];
    }
#pragma unroll
    for (int k = 0; k < 9; ++k) Mg[k * NU + t] = acc[k];
}

// ---------------------------------------------------------------------------
// Kernel 7: AB = msg@Y + Y@msg, decompose, normalize -> R (compressed)
// ---------------------------------------------------------------------------
__device__ __forceinline__ void expand33(const float c[9], float M[3][3]) {
    // c: i0, a01,a02,a12, s00,s11,s01,s02,s12
    M[0][0] = c[0] + c[4];  M[0][1] = c[1] + c[6];  M[0][2] = c[2] + c[7];
    M[1][0] = -c[1] + c[6]; M[1][1] = c[0] + c[5];  M[1][2] = c[3] + c[8];
    M[2][0] = -c[2] + c[7]; M[2][1] = -c[3] + c[8]; M[2][2] = c[0] - c[4] - c[5];
}

__global__ void ab_decompose_kernel(const float* __restrict__ Mg,
                                    const float* __restrict__ Q,
                                    float* __restrict__ R, int NU) {
    int t = blockIdx.x * blockDim.x + threadIdx.x;
    if (t >= NU) return;
    float mc[9], yc[9];
#pragma unroll
    for (int k = 0; k < 9; ++k) { mc[k] = Mg[k * NU + t]; yc[k] = Q[k * NU + t]; }
    float M[3][3], Y[3][3], AB[3][3];
    expand33(mc, M);
    expand33(yc, Y);
#pragma unroll
    for (int i = 0; i < 3; ++i)
#pragma unroll
        for (int j = 0; j < 3; ++j) {
            float s = 0.f;
#pragma unroll
            for (int k = 0; k < 3; ++k) s += M[i][k] * Y[k][j] + Y[i][k] * M[k][j];
            AB[i][j] = s;
        }
    float nrm = 0.f;
#pragma unroll
    for (int i = 0; i < 3; ++i)
#pragma unroll
        for (int j = 0; j < 3; ++j) nrm += AB[i][j] * AB[i][j];
    float inv = 1.f / (nrm + 1.f);
    float i0  = (AB[0][0] + AB[1][1] + AB[2][2]) * (1.f / 3.f);
    R[0 * NU + t] = i0 * inv;
    R[1 * NU + t] = 0.5f * (AB[0][1] - AB[1][0]) * inv;
    R[2 * NU + t] = 0.5f * (AB[0][2] - AB[2][0]) * inv;
    R[3 * NU + t] = 0.5f * (AB[1][2] - AB[2][1]) * inv;
    R[4 * NU + t] = (AB[0][0] - i0) * inv;
    R[5 * NU + t] = (AB[1][1] - i0) * inv;
    R[6 * NU + t] = 0.5f * (AB[0][1] + AB[1][0]) * inv;
    R[7 * NU + t] = 0.5f * (AB[0][2] + AB[2][0]) * inv;
    R[8 * NU + t] = 0.5f * (AB[1][2] + AB[2][1]) * inv;
}

// ---------------------------------------------------------------------------
// Kernel 9: out = X_norm + dX + dX @ dX
// ---------------------------------------------------------------------------
__global__ void final_kernel(const float* __restrict__ Dx,
                             const float* __restrict__ P,
                             float* __restrict__ out, int NU) {
    int t = blockIdx.x * blockDim.x + threadIdx.x;
    if (t >= NU) return;
    float dc[9], pc[9];
#pragma unroll
    for (int k = 0; k < 9; ++k) { dc[k] = Dx[k * NU + t]; pc[k] = P[k * NU + t]; }
    float D[3][3], Xn[3][3];
    expand33(dc, D);
    expand33(pc, Xn);
    float* o = out + (size_t)t * 9;
#pragma unroll
    for (int i = 0; i < 3; ++i)
#pragma unroll
        for (int j = 0; j < 3; ++j) {
            float s = Xn[i][j] + D[i][j];
#pragma unroll
            for (int k = 0; k < 3; ++k) s += D[i][k] * D[k][j];
            o[i * 3 + j] = s;
        }
}

// ---------------------------------------------------------------------------
static inline void launch_chanlin(const float* T, const float* W0,
                                  const float* W1, const float* W2, float* O,
                                  int Nn, hipStream_t stream) {
    int mtiles = Nn / 16;
    int g5 = mtiles / 5;       // 5 M-tiles per wave: exact for Nn=10000 (625=5^4)
    int tail = mtiles % 5;
    if (g5 > 0)
        chanlin_kernel<5><<<9 * g5 * 4, 32, 0, stream>>>(T, W0, W1, W2, O, Nn,
                                                         0, g5);
    if (tail > 0)
        chanlin_kernel<1><<<9 * tail * 4, 32, 0, stream>>>(T, W0, W1, W2, O, Nn,
                                                           g5 * 5, tail);
}

extern "C" void kernel_launch(void* const* d_in, const int* in_sizes, int n_in,
                              void* d_out, int out_size, void* d_ws, size_t ws_size,
                              hipStream_t stream) {
    const int*   eidx = (const int*)d_in[0];
    const float* ew   = (const float*)d_in[1];
    const float* attr = (const float*)d_in[2];
    const float* X    = (const float*)d_in[3];
    const float* Ws0  = (const float*)d_in[4];
    const float* bs0  = (const float*)d_in[5];
    const float* Ws1  = (const float*)d_in[6];
    const float* bs1  = (const float*)d_in[7];
    const float* Ws2  = (const float*)d_in[8];
    const float* bs2  = (const float*)d_in[9];
    const float* Wt0  = (const float*)d_in[10];
    const float* Wt1  = (const float*)d_in[11];
    const float* Wt2  = (const float*)d_in[12];
    const float* Wt3  = (const float*)d_in[13];
    const float* Wt4  = (const float*)d_in[14];
    const float* Wt5  = (const float*)d_in[15];

    int E  = in_sizes[1];
    int Nn = in_sizes[3] / (UNITS * 9);
    int NU = Nn * UNITS;

    float* P  = (float*)d_ws;
    float* Q  = P  + (size_t)9 * NU;
    float* Mg = Q  + (size_t)9 * NU;
    float* R  = Mg + (size_t)9 * NU;
    float* Dx = R  + (size_t)9 * NU;
    float* EA = Dx + (size_t)9 * NU;               // [E,192]
    int* cnt    = (int*)(EA + (size_t)E * 192);
    int* rowptr = cnt + Nn;
    int* cursor = rowptr + (Nn + 1);
    int* elist  = cursor + Nn;

    // 1) normalize + decompose
    node_prep_kernel<<<(NU + 255) / 256, 256, 0, stream>>>(X, P, NU);
    // 2) channel linear with Wt0..2 (WMMA)
    launch_chanlin(P, Wt0, Wt1, Wt2, Q, Nn, stream);
    // 3) CSR build by src
    zero_int_kernel<<<(Nn + 255) / 256, 256, 0, stream>>>(cnt, Nn);
    hist_kernel<<<(E + 255) / 256, 256, 0, stream>>>(eidx, cnt, E);
    scan_kernel<<<1, 1024, 0, stream>>>(cnt, rowptr, cursor, Nn);
    scatter_kernel<<<(E + 255) / 256, 256, 0, stream>>>(eidx, cursor, elist, E);
    // 4) edge MLP (WMMA) -> EA
    int etiles = (E + 31) / 32;
    edge_mlp_kernel<<<(etiles + 3) / 4, 128, 0, stream>>>(
        ew, attr, Ws0, bs0, Ws1, bs1, Ws2, bs2, EA, E);
    // 5) atomic-free message gather -> Mg
    message_gather_kernel<<<(NU + 255) / 256, 256, 0, stream>>>(
        eidx, rowptr, elist, EA, Q, Mg, E, NU);
    // 6) AB = msg@Y + Y@msg, decompose + normalize
    ab_decompose_kernel<<<(NU + 255) / 256, 256, 0, stream>>>(Mg, Q, R, NU);
    // 7) channel linear with Wt3..5 (WMMA)
    launch_chanlin(R, Wt3, Wt4, Wt5, Dx, Nn, stream);
    // 8) out = X_norm + dX + dX@dX
    final_kernel<<<(NU + 255) / 256, 256, 0, stream>>>(Dx, P, (float*)d_out, NU);

    (void)n_in; (void)out_size; (void)ws_size;
}